// GATLayer_61572651155555
// MI455X (gfx1250) — compile-verified
//
#include <hip/hip_runtime.h>
#include <hip/hip_bf16.h>
#include <cstdint>

// GAT layer for MI455X (gfx1250, wave32).
//   h = X @ W^T + b            (dense GEMM -> V_WMMA_F32_16X16X4_F32)
//   logit[e,hd] = leakyrelu(h[src]*a_l + h[dst]*a_r)
//   softmax over out-edges per (src,row); empty rows -> uniform = column mean
//   out[i] = sum_j p(i,j) h[j]

typedef float v2f __attribute__((ext_vector_type(2)));
typedef float v8f __attribute__((ext_vector_type(8)));

#define C_IN 128
#define HDS  4
#define CH   32
#define OUTC 128   // HDS*CH

// ---- order-preserving float<->uint encoding for atomicMax on floats ----
__device__ __forceinline__ unsigned fenc(float f) {
  unsigned u = __float_as_uint(f);
  return (u & 0x80000000u) ? ~u : (u | 0x80000000u);
}
__device__ __forceinline__ float fdec(unsigned u) {
  return (u & 0x80000000u) ? __uint_as_float(u ^ 0x80000000u)
                           : __uint_as_float(~u);
}

// ---------------- init: out=0, rowsum=0, rowmax=-inf ----------------
__global__ void k_init(float* __restrict__ out, unsigned* __restrict__ rowmax,
                       float* __restrict__ rowsum, int N) {
  int idx = blockIdx.x * blockDim.x + threadIdx.x;
  if (idx < N * OUTC) out[idx] = 0.0f;
  if (idx < N * HDS) {
    rowmax[idx] = fenc(-__builtin_inff());
    rowsum[idx] = 0.0f;
  }
}

// ---------------- projection GEMM: h = X @ W^T + b ------------------
// Block = 256 threads = 8 waves. Block computes 16 rows x 128 cols.
// Wave w computes the 16x16 tile at cols [16w, 16w+16).
// A (16x4 f32) per ISA layout: lane<16 holds (M=lane, K=k0+{0,1}),
//                              lane>=16 holds (M=lane-16, K=k0+{2,3}).
// B (4x16)  mirrored: lane<16 -> K=k0+{0,1}, lane>=16 -> K=k0+{2,3}, N=lane&15.
// C/D (16x16 f32): VGPR r -> (M = r + 8*(lane>=16), N = lane&15).
__global__ __launch_bounds__(256) void k_proj(const float* __restrict__ X,
                                              const float* __restrict__ W,
                                              const float* __restrict__ bias,
                                              float* __restrict__ h, int N) {
  __shared__ float sA[16 * C_IN];  // 8 KB tile of X

  const int tid = threadIdx.x;
  const int m0  = blockIdx.x * 16;

  // cooperative stage of the 16x128 X tile (512 float4, 2 per thread)
  const float4* g  = (const float4*)(X + (size_t)m0 * C_IN);
  float4*       s4 = (float4*)sA;
  s4[tid]       = g[tid];
  s4[tid + 256] = g[tid + 256];
  __syncthreads();

  const int wave = tid >> 5;
  const int lane = tid & 31;
  const int o0   = wave * 16;
  const int mrow = lane & 15;          // M (for A) / N (for B,C)
  const int ksel = (lane >> 4) * 2;    // 0 or 2: K sub-select per half-wave

  // B[k, o] = W[o, k]; this lane always reads W row (o0+mrow)
  const float* wrow = W + (size_t)(o0 + mrow) * C_IN;
  const float* arow = sA + mrow * C_IN;

  v8f acc = {};
#pragma unroll 8
  for (int k0 = 0; k0 < C_IN; k0 += 4) {
    v2f afrag = *(const v2f*)(arow + k0 + ksel);   // ds_load_b64
    v2f bfrag = *(const v2f*)(wrow + k0 + ksel);   // global_load_b64
    acc = __builtin_amdgcn_wmma_f32_16x16x4_f32(
        /*neg_a=*/false, afrag, /*neg_b=*/false, bfrag,
        /*c_mod=*/(short)0, acc, /*reuse_a=*/false, /*reuse_b=*/false);
  }

  const int   colg  = o0 + mrow;
  const float bb    = bias[colg];
  const int   rbase = m0 + ((lane >> 4) << 3);  // +8 for upper half-wave
#pragma unroll
  for (int r = 0; r < 8; ++r) {
    h[(size_t)(rbase + r) * OUTC + colg] = acc[r] + bb;
  }
}

// ---------------- column mean of h (for empty softmax rows) ---------
__global__ __launch_bounds__(256) void k_mean(const float* __restrict__ h,
                                              float* __restrict__ hmean, int N) {
  __shared__ float red[256];
  const int col = blockIdx.x;
  float s = 0.0f;
  for (int i = threadIdx.x; i < N; i += 256) s += h[(size_t)i * OUTC + col];
  red[threadIdx.x] = s;
  __syncthreads();
  for (int off = 128; off > 0; off >>= 1) {
    if (threadIdx.x < off) red[threadIdx.x] += red[threadIdx.x + off];
    __syncthreads();
  }
  if (threadIdx.x == 0) hmean[col] = red[0] / (float)N;
}

// ---------------- per-(edge,head) logits + row max ------------------
__global__ __launch_bounds__(256) void k_logit(const float* __restrict__ h,
                                               const float* __restrict__ a,
                                               const int* __restrict__ src,
                                               const int* __restrict__ dst,
                                               float* __restrict__ logit,
                                               unsigned* __restrict__ rowmax,
                                               int E) {
  int idx = blockIdx.x * blockDim.x + threadIdx.x;
  if (idx >= E * HDS) return;
  const int e  = idx >> 2;   // HDS == 4
  const int hd = idx & 3;
  const int i = src[e];
  const int j = dst[e];

  const float4* hi = (const float4*)(h + (size_t)i * OUTC + hd * CH);
  const float4* hj = (const float4*)(h + (size_t)j * OUTC + hd * CH);
  const float4* al = (const float4*)(a + hd * 2 * CH);
  const float4* ar = al + (CH / 4);

  float s = 0.0f;
#pragma unroll
  for (int q = 0; q < CH / 4; ++q) {
    float4 x = hi[q], y = al[q];
    s += x.x * y.x + x.y * y.y + x.z * y.z + x.w * y.w;
    float4 u = hj[q], v = ar[q];
    s += u.x * v.x + u.y * v.y + u.z * v.z + u.w * v.w;
  }
  s = (s >= 0.0f) ? s : 0.2f * s;  // LeakyReLU(0.2)

  logit[idx] = s;
  atomicMax(&rowmax[i * HDS + hd], fenc(s));
}

// ---------------- aggregate: one block per edge (128 thr = H*C) -----
__global__ __launch_bounds__(128) void k_agg(const float* __restrict__ h,
                                             const float* __restrict__ logit,
                                             const unsigned* __restrict__ rowmax,
                                             const int* __restrict__ src,
                                             const int* __restrict__ dst,
                                             float* __restrict__ out,
                                             float* __restrict__ rowsum, int E) {
  const int e  = blockIdx.x;
  const int t  = threadIdx.x;     // 0..127 -> (head, channel)
  const int hd = t >> 5;
  const int i = src[e];
  const int j = dst[e];

  const float l = logit[e * HDS + hd];
  const float m = fdec(rowmax[i * HDS + hd]);
  const float w = expf(l - m);    // <= 1; row with any edge has w_max = 1

  atomicAdd(&out[(size_t)i * OUTC + t], w * h[(size_t)j * OUTC + t]);
  if ((t & 31) == 0) atomicAdd(&rowsum[i * HDS + hd], w);
}

// ---------------- normalize; empty rows -> uniform mean -------------
__global__ void k_norm(float* __restrict__ out, const float* __restrict__ rowsum,
                       const float* __restrict__ hmean, int N) {
  int idx = blockIdx.x * blockDim.x + threadIdx.x;
  if (idx >= N * OUTC) return;
  const int i  = idx >> 7;   // OUTC == 128
  const int t  = idx & 127;
  const int hd = t >> 5;
  const float s = rowsum[i * HDS + hd];
  out[idx] = (s > 0.0f) ? out[idx] / s : hmean[t];
}

// -------------------------------------------------------------------
extern "C" void kernel_launch(void* const* d_in, const int* in_sizes, int n_in,
                              void* d_out, int out_size, void* d_ws, size_t ws_size,
                              hipStream_t stream) {
  const float* X    = (const float*)d_in[0];  // [N, 128]
  const float* W    = (const float*)d_in[1];  // [128, 128]
  const float* bias = (const float*)d_in[2];  // [128]
  const float* a    = (const float*)d_in[3];  // [4, 64]
  const int*   ei   = (const int*)d_in[4];    // [2, E]

  const int N = in_sizes[0] / C_IN;
  const int E = in_sizes[4] / 2;
  const int* src = ei;
  const int* dst = ei + E;
  float* out = (float*)d_out;

  // workspace carve-up (~4.3 MB), segments packed back to back
  char*     ws     = (char*)d_ws;
  float*    h      = (float*)ws;                                    // N*128 f32
  size_t    off    = (size_t)N * OUTC * sizeof(float);
  float*    logit  = (float*)(ws + off);                            // E*4 f32
  off += (size_t)E * HDS * sizeof(float);
  unsigned* rowmax = (unsigned*)(ws + off);                         // N*4 u32
  off += (size_t)N * HDS * sizeof(unsigned);
  float*    rowsum = (float*)(ws + off);                            // N*4 f32
  off += (size_t)N * HDS * sizeof(float);
  float*    hmean  = (float*)(ws + off);                            // 128 f32

  k_init <<<(N * OUTC + 255) / 256, 256, 0, stream>>>(out, rowmax, rowsum, N);
  k_proj <<<N / 16, 256, 0, stream>>>(X, W, bias, h, N);
  k_mean <<<OUTC, 256, 0, stream>>>(h, hmean, N);
  k_logit<<<(E * HDS + 255) / 256, 256, 0, stream>>>(h, a, src, dst, logit, rowmax, E);
  k_agg  <<<E, 128, 0, stream>>>(h, logit, rowmax, src, dst, out, rowsum, E);
  k_norm <<<(N * OUTC + 255) / 256, 256, 0, stream>>>(out, rowsum, hmean, N);
}